// LocalRetinaNet_16930761081210
// MI455X (gfx1250) — compile-verified
//
#include <hip/hip_runtime.h>
#include <stdint.h>

typedef __attribute__((ext_vector_type(2))) float v2f;
typedef __attribute__((ext_vector_type(8))) float v8f;

#define BATCH 16
#define NANCH 193536   // 9*(128*128 + 64*64 + 32*32)
#define L0_END 147456  // 9*16384
#define L1_END 184320  // +9*4096
#define KPAD  1024
#define KOUT  1000

__device__ __forceinline__ void decode_level(int n, int& lvl, int& local) {
    if (n < L0_END)      { lvl = 0; local = n; }
    else if (n < L1_END) { lvl = 1; local = n - L0_END; }
    else                 { lvl = 2; local = n - L1_END; }
}

// ---------------- Kernel 1: sigmoid(cls) -> ws_scores (B, N) ----------------
__global__ void __launch_bounds__(256)
k_sigmoid(const float* __restrict__ c3, const float* __restrict__ c4,
          const float* __restrict__ c5, float* __restrict__ sc) {
    int i = blockIdx.x * blockDim.x + threadIdx.x;
    if (i >= BATCH * NANCH) return;
    int b = i / NANCH, n = i % NANCH;
    int lvl, local; decode_level(n, lvl, local);
    const float* p; int hw;
    if (lvl == 0)      { p = c3; hw = 16384; }
    else if (lvl == 1) { p = c4; hw = 4096; }
    else               { p = c5; hw = 1024; }
    int a = local / hw, pos = local % hw;
    float v = p[((size_t)b * 9 + a) * hw + pos];
    sc[i] = __fdividef(1.0f, 1.0f + __expf(-v));
}

// ------- Kernel 2: per-image top-1000 select + sort + decode boxes ----------
__global__ void __launch_bounds__(1024)
k_topk(const float* __restrict__ sc,
       const float* __restrict__ b3, const float* __restrict__ b4,
       const float* __restrict__ b5,
       float* __restrict__ topk_sc, int* __restrict__ topk_id,
       float* __restrict__ topk_box) {
    __shared__ unsigned hist[4096];
    __shared__ float ssc[KPAD];
    __shared__ int   sid[KPAD];
    __shared__ unsigned s_T, s_nAbove, s_cntA, s_cntT;

    int b = blockIdx.x, tid = threadIdx.x;
    for (int i = tid; i < 4096; i += 1024) hist[i] = 0u;
    ssc[tid] = -1.0f; sid[tid] = 0;
    if (tid == 0) { s_cntA = 0u; s_cntT = 0u; s_nAbove = 0u; s_T = 0u; }
    __syncthreads();

    const float* s = sc + (size_t)b * NANCH;
    // scores are sigmoid outputs (positive), so float bits are order-preserving
    for (int n = tid; n < NANCH; n += 1024)
        atomicAdd(&hist[__float_as_uint(s[n]) >> 20], 1u);
    __syncthreads();

    if (tid == 0) {
        unsigned cum = 0; int T = 0;
        for (int bin = 4095; bin >= 0; --bin) {
            unsigned c = hist[bin];
            if (cum + c >= KOUT) { T = bin; s_nAbove = cum; break; }
            cum += c;
        }
        s_T = (unsigned)T;
    }
    __syncthreads();
    unsigned T = s_T, nAbove = s_nAbove, needT = KOUT - nAbove;

    for (int n = tid; n < NANCH; n += 1024) {
        float v = s[n];
        unsigned bin = __float_as_uint(v) >> 20;
        if (bin > T) {
            unsigned p = atomicAdd(&s_cntA, 1u);
            ssc[p] = v; sid[p] = n;
        } else if (bin == T) {
            unsigned q = atomicAdd(&s_cntT, 1u);
            if (q < needT) { ssc[nAbove + q] = v; sid[nAbove + q] = n; }
        }
    }
    __syncthreads();

    // bitonic sort (descending by score), 1024 elements
    for (int k = 2; k <= KPAD; k <<= 1) {
        for (int j = k >> 1; j > 0; j >>= 1) {
            int ixj = tid ^ j;
            if (ixj > tid) {
                bool up = ((tid & k) == 0);
                float a0 = ssc[tid], a1 = ssc[ixj];
                int   i0 = sid[tid], i1 = sid[ixj];
                if ((a0 < a1) == up) {
                    ssc[tid] = a1; ssc[ixj] = a0;
                    sid[tid] = i1; sid[ixj] = i0;
                }
            }
            __syncthreads();
        }
    }

    // decode this thread's selected anchor
    float v = ssc[tid]; int n = sid[tid];
    int lvl, local; decode_level(n, lvl, local);
    const float* bp; int hw, w, stride; float base;
    if (lvl == 0)      { bp = b3; hw = 16384; w = 128; stride = 8;  base = 32.f; }
    else if (lvl == 1) { bp = b4; hw = 4096;  w = 64;  stride = 16; base = 64.f; }
    else               { bp = b5; hw = 1024;  w = 32;  stride = 32; base = 128.f; }
    int a = local / hw, pos = local % hw;
    int y = pos / w, x = pos % w;
    const float scl[3] = {1.0f, 1.2599210499f, 1.5874010520f};
    const float sr[3]  = {0.70710678119f, 1.0f, 1.41421356237f};
    float size = base * scl[a / 3];
    float aw = size * sr[a % 3], ah = size / sr[a % 3];
    float cx = (x + 0.5f) * (float)stride, cy = (y + 0.5f) * (float)stride;
    size_t di = (((size_t)b * 36 + a * 4) * hw) + pos;
    float tx = bp[di], ty = bp[di + hw], tw = bp[di + 2 * (size_t)hw], th = bp[di + 3 * (size_t)hw];
    float gx = tx * aw + cx, gy = ty * ah + cy;
    float gw = aw * __expf(tw), gh = ah * __expf(th);
    float x1 = gx - 0.5f * gw, y1 = gy - 0.5f * gh;
    float x2 = gx + 0.5f * gw, y2 = gy + 0.5f * gh;
    if (v < 0.f) { x1 = y1 = x2 = y2 = 0.f; }  // padding rows

    size_t o = (size_t)b * KPAD + tid;
    topk_sc[o] = v; topk_id[o] = n;
    float* bo = topk_box + o * 4;
    bo[0] = x1; bo[1] = y1; bo[2] = x2; bo[3] = y2;
}

// -------- Kernel 3: adjacency (WMMA outer-sum) + greedy NMS + output --------
__global__ void __launch_bounds__(1024)
k_nms(const float* __restrict__ topk_sc, const float* __restrict__ topk_box,
      unsigned* __restrict__ adj, float* __restrict__ out) {
    __shared__ float x1s[KPAD], y1s[KPAD], x2s[KPAD], y2s[KPAD], scs[KPAD], ars[KPAD];
    __shared__ unsigned keepw[32];

    int b = blockIdx.x, tid = threadIdx.x;
    int lane = tid & 31, wave = tid >> 5;

    // ---- async stage boxes (AoS global) -> SoA LDS via CDNA5 async loads ----
    {
        unsigned long long g0 =
            (unsigned long long)(uintptr_t)(topk_box + ((size_t)b * KPAD + tid) * 4);
        unsigned long long g4 =
            (unsigned long long)(uintptr_t)(topk_sc + ((size_t)b * KPAD + tid));
        unsigned d0 = (unsigned)(uintptr_t)&x1s[tid];
        unsigned d1 = (unsigned)(uintptr_t)&y1s[tid];
        unsigned d2 = (unsigned)(uintptr_t)&x2s[tid];
        unsigned d3 = (unsigned)(uintptr_t)&y2s[tid];
        unsigned d4 = (unsigned)(uintptr_t)&scs[tid];
        asm volatile("global_load_async_to_lds_b32 %0, %1, off" :: "v"(d0), "v"(g0) : "memory");
        asm volatile("global_load_async_to_lds_b32 %0, %1, off" :: "v"(d1), "v"(g0 + 4) : "memory");
        asm volatile("global_load_async_to_lds_b32 %0, %1, off" :: "v"(d2), "v"(g0 + 8) : "memory");
        asm volatile("global_load_async_to_lds_b32 %0, %1, off" :: "v"(d3), "v"(g0 + 12) : "memory");
        asm volatile("global_load_async_to_lds_b32 %0, %1, off" :: "v"(d4), "v"(g4) : "memory");
        asm volatile("s_wait_asynccnt 0" ::: "memory");
    }
    __syncthreads();

    ars[tid] = (x2s[tid] - x1s[tid]) * (y2s[tid] - y1s[tid]);
    unsigned kb = (unsigned)__ballot(scs[tid] > 0.05f);
    if (lane == 0) keepw[wave] = kb;

    unsigned* arow = adj + (size_t)b * KPAD * 32;
    __syncthreads();

    // ---- adjacency bits: each wave owns 16-row strips (ti = wave, wave+32),
    //      sweeps tj tiles in pairs; every adjacency word stored exactly once ----
    bool lo = lane < 16;
    int l16 = lane & 15;
    bool hi_store = (l16 == 0);            // lanes 0 and 16 store composed words
    int rsel = (lane >> 4) << 3;           // 0 for lanes 0-15, 8 for lanes 16-31

    for (int ti = wave; ti < 64; ti += 32) {
        int i0 = ti * 16;

        // A (16x4 f32): col0 = area_a, col1 = 1, cols 2..3 = 0 (hoisted per strip).
        // B rows 2..3 need no masking: they multiply A cols 2..3 == 0.
        v2f A;
        A.x = lo ? ars[i0 + l16] : 0.f;   // K=0 (lanes 0-15) / K=2 (lanes 16-31)
        A.y = lo ? 1.f : 0.f;             // K=1 / K=3

        // hoist this lane's 8 accumulator-layout row boxes into registers
        float rx1[8], ry1[8], rx2[8], ry2[8];
        #pragma unroll
        for (int v = 0; v < 8; ++v) {
            int ii = i0 + v + rsel;
            rx1[v] = x1s[ii]; ry1[v] = y1s[ii];
            rx2[v] = x2s[ii]; ry2[v] = y2s[ii];
        }

        // one 16x16 tile: D = area_a[i] + area_b[j] via WMMA, then ballot masks
        auto tile_masks = [&](int j0, unsigned (&mv)[8]) {
            v2f Bm;
            Bm.x = 1.f;               // row0 = 1   (rows 2/3 don't matter)
            Bm.y = ars[j0 + l16];     // row1 = area_b
            v8f C = {0.f, 0.f, 0.f, 0.f, 0.f, 0.f, 0.f, 0.f};
            v8f D = __builtin_amdgcn_wmma_f32_16x16x4_f32(false, A, false, Bm,
                                                          (short)0, C, false, false);
            int jj = j0 + l16;
            float jx1 = x1s[jj], jy1 = y1s[jj], jx2 = x2s[jj], jy2 = y2s[jj];
            #pragma unroll
            for (int v = 0; v < 8; ++v) {
                float w = fminf(rx2[v], jx2) - fmaxf(rx1[v], jx1);
                float h = fminf(ry2[v], jy2) - fmaxf(ry1[v], jy1);
                w = fmaxf(w, 0.f); h = fmaxf(h, 0.f);
                float inter = w * h;
                // iou > 0.5  <=>  inter > 0.5 * union  (union = D - inter + eps > 0)
                bool pred = inter > 0.5f * (D[v] - inter + 1e-8f);
                mv[v] = (unsigned)__ballot(pred);
            }
        };

        for (int tjp = 0; tjp < 32; ++tjp) {
            unsigned me[8], mo[8];
            tile_masks(tjp * 32,      me);   // even tile: bits [15:0] of word tjp
            tile_masks(tjp * 32 + 16, mo);   // odd tile:  bits [31:16]
            #pragma unroll
            for (int v = 0; v < 8; ++v) {
                unsigned wl = (me[v] & 0xFFFFu) | ((mo[v] & 0xFFFFu) << 16);
                unsigned wh = (me[v] >> 16)     | ((mo[v] >> 16) << 16);
                if (hi_store) {
                    unsigned dat = (rsel == 0) ? wl : wh;     // v_cndmask on uniforms
                    arow[(i0 + v + rsel) * 32 + tjp] = dat;
                }
            }
        }
    }
    __threadfence();
    __syncthreads();

    // ---- sequential greedy pass over bitmask rows (wave 0 only) ----
    if (wave == 0) {
        unsigned kw = keepw[lane];
        for (int i = 0; i < KPAD; ++i) {
            int wi = i >> 5;
            unsigned wb = (unsigned)__shfl((int)kw, wi);
            if (wb & (1u << (i & 31))) {
                unsigned a = arow[i * 32 + lane];
                if (lane < wi)       a = 0u;
                else if (lane == wi) a &= ~((2u << (i & 31)) - 1u); // only j > i
                kw &= ~a;
            }
        }
        keepw[lane] = kw;
    }
    __syncthreads();

    if (tid < KOUT) {
        float kf = ((keepw[tid >> 5] >> (tid & 31)) & 1u) ? 1.f : 0.f;
        float* o = out + ((size_t)b * KOUT + tid) * 5;
        o[0] = x1s[tid] * kf; o[1] = y1s[tid] * kf;
        o[2] = x2s[tid] * kf; o[3] = y2s[tid] * kf;
        o[4] = scs[tid] * kf;
    }
}

extern "C" void kernel_launch(void* const* d_in, const int* in_sizes, int n_in,
                              void* d_out, int out_size, void* d_ws, size_t ws_size,
                              hipStream_t stream) {
    (void)in_sizes; (void)n_in; (void)out_size; (void)ws_size;
    const float* c3 = (const float*)d_in[0];
    const float* c4 = (const float*)d_in[1];
    const float* c5 = (const float*)d_in[2];
    const float* b3 = (const float*)d_in[3];
    const float* b4 = (const float*)d_in[4];
    const float* b5 = (const float*)d_in[5];
    float* out = (float*)d_out;

    char* ws = (char*)d_ws;
    size_t off = 0;
    float* ws_scores = (float*)(ws + off); off += (size_t)BATCH * NANCH * sizeof(float);
    float* ws_tsc    = (float*)(ws + off); off += (size_t)BATCH * KPAD * sizeof(float);
    int*   ws_tid    = (int*)  (ws + off); off += (size_t)BATCH * KPAD * sizeof(int);
    float* ws_tbox   = (float*)(ws + off); off += (size_t)BATCH * KPAD * 4 * sizeof(float);
    unsigned* ws_adj = (unsigned*)(ws + off);

    k_sigmoid<<<(BATCH * NANCH) / 256, 256, 0, stream>>>(c3, c4, c5, ws_scores);
    k_topk<<<BATCH, 1024, 0, stream>>>(ws_scores, b3, b4, b5, ws_tsc, ws_tid, ws_tbox);
    k_nms<<<BATCH, 1024, 0, stream>>>(ws_tsc, ws_tbox, ws_adj, out);
}